// SeesawLossWithLogits_3049426780528
// MI455X (gfx1250) — compile-verified
//
#include <hip/hip_runtime.h>

typedef __attribute__((ext_vector_type(2))) float v2f;
typedef __attribute__((ext_vector_type(8))) float v8f;

#define NUMC 128

// Seesaw loss, collapsed to per-sample gathered dot product:
//   den[n]  = sum_j exp(x[n,j]-max_n) * s[t_n, j]     (s[t,t]==1 makes terms cancel)
//   loss[n] = -log( e_t / (den + 1e-6) + 1e-6 ),  output = mean(loss)
// A wave processes 16 samples; den for the 16 samples = diag(E(16x128) x Sg^T(128x16)),
// computed with 32 chained v_wmma_f32_16x16x4_f32 (full fp32, matches reference).
__global__ __launch_bounds__(256) void seesaw_wmma_kernel(
    const float* __restrict__ logits, const int* __restrict__ targets,
    const float* __restrict__ s, float* __restrict__ out, int N, float invN)
{
    // s transposed into LDS: s_cm[c*128 + i] = s[i*128 + c]  (64 KB)
    __shared__ float s_cm[NUMC * NUMC];
    const int tid = threadIdx.x;
    for (int idx = tid; idx < NUMC * NUMC; idx += 256) {
        int i = idx & (NUMC - 1);   // row of s   (consecutive lanes -> distinct banks on write)
        int c = idx >> 7;           // col of s
        s_cm[c * NUMC + i] = s[i * NUMC + c];
    }
    __syncthreads();

    const int lane = tid & 31;
    const int wid  = tid >> 5;
    const int m    = lane & 15;   // sample-in-group == WMMA A row
    const int h    = lane >> 4;   // which K-half this lane feeds
    const int groups     = N >> 4;
    const int totalWaves = gridDim.x * 8;
    const int waveGid    = blockIdx.x * 8 + wid;

    float lsum = 0.0f;

    for (int grp = waveGid; grp < groups; grp += totalWaves) {
        const int base = grp << 4;
        const int row  = base + m;
        const float* rowp = logits + (size_t)row * NUMC;
        const int t = targets[row];

        // Lane owns columns {4q+2h, 4q+2h+1}, q=0..31 : 32 x 8B loads, row read once.
        float2 vals[32];
        #pragma unroll
        for (int q = 0; q < 32; ++q)
            vals[q] = *(const float2*)(rowp + 4 * q + 2 * h);

        // full-row max: local 64 elems, then swap with partner half-lane
        float mx = -3.402823466e+38f;
        #pragma unroll
        for (int q = 0; q < 32; ++q)
            mx = fmaxf(mx, fmaxf(vals[q].x, vals[q].y));
        mx = fmaxf(mx, __shfl_xor(mx, 16, 32));

        // e = exp(x - max), kept in registers as the WMMA A operand stream
        #pragma unroll
        for (int q = 0; q < 32; ++q) {
            vals[q].x = __expf(vals[q].x - mx);
            vals[q].y = __expf(vals[q].y - mx);
        }

        // exp at the target column
        const float et = __expf(rowp[t] - mx);

        // acc(16x16) += A(16x4) x B(4x16) over 32 K-steps.
        // A lane layout: row m=lane%16, K = g + 2h  -> exactly vals[q].{x,y}
        // B lane layout (mirror): col n=lane%16, K = g + 2h -> s[t_n, 4q+2h+g]
        v8f acc = {0.f, 0.f, 0.f, 0.f, 0.f, 0.f, 0.f, 0.f};
        #pragma unroll
        for (int q = 0; q < 32; ++q) {
            const int c0 = 4 * q + 2 * h;
            v2f a; a.x = vals[q].x;              a.y = vals[q].y;
            v2f b; b.x = s_cm[c0 * NUMC + t];    b.y = s_cm[(c0 + 1) * NUMC + t];
            acc = __builtin_amdgcn_wmma_f32_16x16x4_f32(
                false, a, false, b, (short)0, acc, false, false);
        }

        // diagonal of C/D layout: lane holds N=lane%16, M = v + 8*(lane>>4)
        // -> diag element lives at v = m - 8h when 0 <= v < 8
        const int v = m - (h << 3);
        float den = (v == 0) ? acc[0] : (v == 1) ? acc[1] : (v == 2) ? acc[2]
                  : (v == 3) ? acc[3] : (v == 4) ? acc[4] : (v == 5) ? acc[5]
                  : (v == 6) ? acc[6] : acc[7];

        const float sigma = et / (den + 1e-6f);
        const float l = -__logf(sigma + 1e-6f);
        lsum += (v >= 0 && v < 8) ? l : 0.0f;
    }

    // wave tree reduction, one atomic per wave (<= 4096 total)
    #pragma unroll
    for (int off = 16; off > 0; off >>= 1)
        lsum += __shfl_xor(lsum, off, 32);
    if (lane == 0) atomicAdd(out, lsum * invN);
}

extern "C" void kernel_launch(void* const* d_in, const int* in_sizes, int n_in,
                              void* d_out, int out_size, void* d_ws, size_t ws_size,
                              hipStream_t stream) {
    const float* logits  = (const float*)d_in[0];
    const int*   targets = (const int*)d_in[1];
    const float* s       = (const float*)d_in[2];
    float* out = (float*)d_out;
    const int N = in_sizes[1];

    hipMemsetAsync(out, 0, sizeof(float), stream);

    const int groups = N >> 4;
    int grid = (groups + 7) / 8;
    if (grid > 512) grid = 512;
    if (grid < 1)   grid = 1;
    seesaw_wmma_kernel<<<grid, 256, 0, stream>>>(logits, targets, s, out, N,
                                                 1.0f / (float)N);
}